// LogicMP_84301618085891
// MI455X (gfx1250) — compile-verified
//
#include <hip/hip_runtime.h>

// LogicMP belief propagation, B=2, N=2048, NSTEP=5.
// Bandwidth-bound (~400MB total traffic -> ~17us @ 23.3TB/s). d = F1-F0 staged
// in the spare half of the F output region (epilogue overwrites it last).
// Col-dot reduction done with V_WMMA_F32_16X16X4_F32 (A = ones => D[m,n] =
// sum_k B[k,n]; robust to K-slot layout), row-dot via registers + shfl_xor.

#define NN   2048
#define BB   2
#define BN   (BB*NN)          // 4096
#define YS   16               // y-slices per row band
#define LOG2E 1.44269504088896340736f

typedef float v2f __attribute__((ext_vector_type(2)));
typedef float v8f __attribute__((ext_vector_type(8)));

// ws layout (floats): AS[7][BN] | CS[7][BN] | rowdot[BN] | coldot[BN]
// AS slot (k+1) holds qS_k[...,1] for k = -1..5 (slot 0 is the zero "step -1").

__device__ __forceinline__ float sigm(float t) {
  // 1/(1+exp(-t))
  float e = __builtin_amdgcn_exp2f(-t * LOG2E);
  return __builtin_amdgcn_rcpf(1.0f + e);
}

__global__ __launch_bounds__(256) void k_init(const float* __restrict__ lS,
                                              float* __restrict__ ws) {
  int x = blockIdx.x * blockDim.x + threadIdx.x;
  if (x >= BN) return;
  float* AS = ws;
  float* CS = ws + 7 * BN;
  float* rowdot = ws + 14 * BN;
  float* coldot = ws + 15 * BN;
  float s0 = lS[2 * x], s1 = lS[2 * x + 1];
  float a = sigm(s1 - s0);
  AS[x] = 0.f;  CS[x] = 0.f;                 // step -1 (zeros)
  AS[BN + x] = a; CS[BN + x] = 1.f - a;      // step 0
  rowdot[x] = 0.f; coldot[x] = 0.f;
}

// d[b,x,y] = F[...,1] - F[...,0]; one float4 (2 pairs) per thread.
__global__ __launch_bounds__(256) void k_d(const float4* __restrict__ F,
                                           float2* __restrict__ dmat) {
  long i = (long)blockIdx.x * blockDim.x + threadIdx.x;
  if (i >= (long)BB * NN * NN / 2) return;
  float4 f = F[i];
  dmat[i] = make_float2(f.y - f.x, f.w - f.z);
}

// One wave handles a 16-row band x 128-col slice; 16x16 tiles.
// B-operand element (lane,reg r,chunk c): row x0+4c+r+2*hi, col yt+lo.
__global__ __launch_bounds__(256) void k_step(const float* __restrict__ dmat,
                                              float* __restrict__ ws,
                                              const float* __restrict__ wts,
                                              int t) {
  const float w1 = wts[1];
  const float* AS = ws;
  const float* CS = ws + 7 * BN;
  float* rowdot = ws + 14 * BN;
  float* coldot = ws + 15 * BN;
  const float* aP = AS + t * BN;        // qS_{t-1}[...,1]   (zeros for t=0)
  const float* cP = CS + t * BN;
  const float* aC = AS + (t + 1) * BN;  // qS_t[...,1]
  const float* cC = CS + (t + 1) * BN;

  const int lane = threadIdx.x & 31;
  const int wv   = threadIdx.x >> 5;
  const int job  = blockIdx.x * 8 + wv;          // 0 .. B*128*YS-1
  const int b    = job / (128 * YS);
  const int rem  = job % (128 * YS);
  const int x0   = (rem / YS) * 16;
  const int y0   = (rem % YS) * (NN / YS);
  const int hi = lane >> 4, lo = lane & 15;

  float w1aPL[8], aCur[8], rowp[8];
  int xr[8];
#pragma unroll
  for (int c = 0; c < 4; c++)
#pragma unroll
    for (int r = 0; r < 2; r++) {
      int i = c * 2 + r;
      int x = x0 + 4 * c + r + 2 * hi;
      xr[i] = x;
      w1aPL[i] = (w1 * aP[b * NN + x]) * LOG2E;
      aCur[i]  = aC[b * NN + x];
      rowp[i]  = 0.f;
    }

  v2f ones; ones.x = 1.0f; ones.y = 1.0f;        // A = ones(16x4), any layout
  const float* drow = dmat + (long)b * NN * NN;

#pragma unroll 2
  for (int ti = 0; ti < (NN / YS) / 16; ti++) {
    int yt = y0 + ti * 16;
    float cPy = cP[b * NN + yt + lo];
    float cCy = cC[b * NN + yt + lo];
    v8f acc = {0.f, 0.f, 0.f, 0.f, 0.f, 0.f, 0.f, 0.f};
#pragma unroll
    for (int c = 0; c < 4; c++) {
      v2f Bv;
#pragma unroll
      for (int r = 0; r < 2; r++) {
        int i = c * 2 + r;
        float dv = drow[(long)xr[i] * NN + yt + lo];
        float ee = __builtin_amdgcn_exp2f(w1aPL[i] * cPy - dv * LOG2E);
        float q  = __builtin_amdgcn_rcpf(1.0f + ee);   // qF[...,1]
        rowp[i] += q * cCy;                            // row-dot partial
        Bv[r] = q * aCur[i];                           // col-dot operand
      }
      // D[m,n] += sum_k B[k,n]  (A = ones)
      acc = __builtin_amdgcn_wmma_f32_16x16x4_f32(
          false, ones, false, Bv, (short)0, acc, false, false);
    }
    // C/D layout: reg0, lanes 0-15 => (M=0, N=lo) = coldot partial for yt+lo
    if (hi == 0) atomicAdd(&coldot[b * NN + yt + lo], acc[0]);
  }

  // reduce row partials across the 16 lanes of each half (lo dimension)
#pragma unroll
  for (int i = 0; i < 8; i++) {
    float v = rowp[i];
    v += __shfl_xor(v, 1, 32);
    v += __shfl_xor(v, 2, 32);
    v += __shfl_xor(v, 4, 32);
    v += __shfl_xor(v, 8, 32);
    if (lo == 0) atomicAdd(&rowdot[b * NN + xr[i]], v);
  }
}

__global__ __launch_bounds__(256) void k_update(const float* __restrict__ lS,
                                                const float* __restrict__ lC,
                                                float* __restrict__ ws,
                                                const float* __restrict__ wts,
                                                float* __restrict__ outS,
                                                float* __restrict__ outC,
                                                int t) {
  int x = blockIdx.x * blockDim.x + threadIdx.x;
  if (x >= BN) return;
  const float w0 = wts[0], w1 = wts[1];
  float* AS = ws;
  float* CS = ws + 7 * BN;
  float* rowdot = ws + 14 * BN;
  float* coldot = ws + 15 * BN;
  float aPm1 = AS[t * BN + x];         // qS_{t-1} (zero slot when t==0)
  float aT   = AS[(t + 1) * BN + x];   // qS_t
  float s0l = lS[2 * x], s1l = lS[2 * x + 1];
  float c0l = lC[2 * x], c1l = lC[2 * x + 1];
  float qC0 = sigm(c0l - (c1l + w0 * aPm1));        // softmax(cur_C_t)[0]
  float s0 = s0l + w0 * qC0 + w1 * rowdot[x];       // cur_S_{t+1}
  float s1 = s1l + w1 * coldot[x];
  outS[2 * x] = s0; outS[2 * x + 1] = s1;
  outC[2 * x] = c0l; outC[2 * x + 1] = c1l + w0 * aT;   // cur_C_{t+1}
  float a = sigm(s1 - s0);                               // qS_{t+1}
  AS[(t + 2) * BN + x] = a;
  CS[(t + 2) * BN + x] = 1.f - a;
  rowdot[x] = 0.f; coldot[x] = 0.f;
}

// cur_F_5 = logits_F + msg_F(qS_4): slot0 += w1*a4[x]*c4[y], slot1 unchanged.
__global__ __launch_bounds__(256) void k_final(const float4* __restrict__ F,
                                               float4* __restrict__ outF,
                                               const float* __restrict__ ws,
                                               const float* __restrict__ wts) {
  long i = (long)blockIdx.x * blockDim.x + threadIdx.x;   // float4 = 2 pairs
  if (i >= (long)BB * NN * NN / 2) return;
  const float w1 = wts[1];
  const float* A4 = ws + 5 * BN;            // qS_4 slot
  const float* C4 = ws + 7 * BN + 5 * BN;
  long p = i * 2;                           // first (x,y) pair index
  long b = p / ((long)NN * NN);
  long r = p - b * (long)NN * NN;
  int x = (int)(r / NN), y = (int)(r % NN); // y even; y+1 in same row
  float a4 = A4[b * NN + x];
  float c40 = C4[b * NN + y], c41 = C4[b * NN + y + 1];
  float4 f = F[i];
  float4 o;
  o.x = f.x + w1 * a4 * c40; o.y = f.y;
  o.z = f.z + w1 * a4 * c41; o.w = f.w;
  outF[i] = o;
}

extern "C" void kernel_launch(void* const* d_in, const int* in_sizes, int n_in,
                              void* d_out, int out_size, void* d_ws, size_t ws_size,
                              hipStream_t stream) {
  const float* lS  = (const float*)d_in[0];
  const float* lC  = (const float*)d_in[1];
  const float* lF  = (const float*)d_in[2];
  const float* wts = (const float*)d_in[3];
  float* outS = (float*)d_out;                 // [B,N,2]
  float* outC = outS + 2 * BN;                 // [B,N,2]
  float* outF = outS + 4 * BN;                 // [B,N,N,2]
  float* ws   = (float*)d_ws;                  // needs 16*BN floats = 256 KB
  float2* dmat = (float2*)outF;                // stage d in F output region

  const long nq4 = (long)BB * NN * NN / 2;     // float4 count over F
  dim3 blk(256);
  dim3 gSmall((BN + 255) / 256);
  dim3 gBig((unsigned)((nq4 + 255) / 256));
  dim3 gStep((BB * 128 * YS) / 8);             // 8 waves per block

  k_init<<<gSmall, blk, 0, stream>>>(lS, ws);
  k_d<<<gBig, blk, 0, stream>>>((const float4*)lF, dmat);
  for (int t = 0; t < 5; t++) {
    k_step<<<gStep, blk, 0, stream>>>((const float*)dmat, ws, wts, t);
    k_update<<<gSmall, blk, 0, stream>>>(lS, lC, ws, wts, outS, outC, t);
  }
  k_final<<<gBig, blk, 0, stream>>>((const float4*)lF, (float4*)outF, ws, wts);
}